// MultiHeadAttention_592705486897
// MI455X (gfx1250) — compile-verified
//
#include <hip/hip_runtime.h>

typedef __attribute__((ext_vector_type(2))) float v2f;
typedef __attribute__((ext_vector_type(8))) float v8f;

#define S_LEN   2048
#define D_MODEL 1024
#define N_HEADS 16
#define D_HEAD  64
#define BATCH   2
#define M_ROWS  (BATCH * S_LEN)       // 4096
#define SCALE   0.125f                // 1/sqrt(dh)
#define MASK_ALPHA (-1e9f)
#define LDP     (S_LEN + 4)           // padded LDS row stride (floats)
#define RED_LD  68                    // padded reduction row stride
#define NTILES  4                     // N register blocking in GEMM

__device__ __forceinline__ v8f wmma_f32(v2f a, v2f b, v8f c) {
  // V_WMMA_F32_16X16X4_F32: D = A(16x4) x B(4x16) + C(16x16), full fp32
  return __builtin_amdgcn_wmma_f32_16x16x4_f32(false, a, false, b, (short)0, c,
                                               false, false);
}

// ---------------------------------------------------------------------------
// GEMM: out(M x 1024) = A(M x 1024) @ W(1024 x 1024) + bias
// 8 waves per block; each wave computes a 16x64 tile (4 WMMA N-tiles sharing
// one A fragment). Block tile = 128 x 64.
// scatter_heads=1: write to (B, H, S, dh) layout; 0: plain row-major.
// ---------------------------------------------------------------------------
__global__ __launch_bounds__(256) void gemm_proj(const float* __restrict__ A,
                                                 const float* __restrict__ W,
                                                 const float* __restrict__ bias,
                                                 float* __restrict__ out,
                                                 int scatter_heads) {
  const int lane = threadIdx.x & 31;
  const int wave = threadIdx.x >> 5;
  const int half = lane >> 4;     // lane group 0..15 vs 16..31
  const int lr   = lane & 15;

  const int row_base = blockIdx.x * 128 + wave * 16;
  const int col_base = blockIdx.y * 64;

  const float* a_ptr = A + (size_t)(row_base + lr) * D_MODEL + half * 2;
  const float* w_ptr = W + (size_t)(half * 2) * D_MODEL + col_base + lr;

  v8f acc0 = {}, acc1 = {}, acc2 = {}, acc3 = {};
  for (int kk = 0; kk < D_MODEL; kk += 4) {
    const v2f a = *(const v2f*)(a_ptr + kk);
    const float* wr0 = w_ptr + (size_t)kk * D_MODEL;
    const float* wr1 = wr0 + D_MODEL;
    v2f b0, b1, b2, b3;
    b0.x = wr0[0];   b0.y = wr1[0];
    b1.x = wr0[16];  b1.y = wr1[16];
    b2.x = wr0[32];  b2.y = wr1[32];
    b3.x = wr0[48];  b3.y = wr1[48];
    acc0 = wmma_f32(a, b0, acc0);
    acc1 = wmma_f32(a, b1, acc1);
    acc2 = wmma_f32(a, b2, acc2);
    acc3 = wmma_f32(a, b3, acc3);
  }

  v8f accs[NTILES] = {acc0, acc1, acc2, acc3};
  #pragma unroll
  for (int nt = 0; nt < NTILES; ++nt) {
    const int col = col_base + nt * 16 + lr;
    const float bv = bias[col];
    if (scatter_heads) {
      const int h = col >> 6, d = col & 63;
      #pragma unroll
      for (int j = 0; j < 8; ++j) {
        const int m = row_base + half * 8 + j;
        const int b = m >> 11;          // m / 2048
        const int s = m & (S_LEN - 1);
        out[(((size_t)(b * N_HEADS + h) * S_LEN) + s) * D_HEAD + d] =
            accs[nt][j] + bv;
      }
    } else {
      #pragma unroll
      for (int j = 0; j < 8; ++j) {
        const int m = row_base + half * 8 + j;
        out[(size_t)m * D_MODEL + col] = accs[nt][j] + bv;
      }
    }
  }
}

// ---------------------------------------------------------------------------
// Fused attention: per (b, h, 16 q-rows):
//   logits (WMMA) -> qkvalues, masked logits -> LDS, softmax -> attn,
//   Z = P @ V (WMMA, split-K across waves) -> z_ws (B,S,D layout)
// ---------------------------------------------------------------------------
__global__ __launch_bounds__(256) void attn_fused(const float* __restrict__ q_ws,
                                                  const float* __restrict__ k_ws,
                                                  const float* __restrict__ v_ws,
                                                  const float* __restrict__ mask,
                                                  float* __restrict__ attn_out,
                                                  float* __restrict__ qk_out,
                                                  float* __restrict__ z_ws) {
  extern __shared__ float smem[];
  float* P   = smem;                  // 16 x LDP masked logits / probs
  float* red = smem + 16 * LDP;       // 2 x 16 x RED_LD split-K partials

  const int lane = threadIdx.x & 31;
  const int wave = threadIdx.x >> 5;
  const int half = lane >> 4;
  const int lr   = lane & 15;

  const int bh     = blockIdx.y;      // b*16 + h
  const int b      = bh >> 4;
  const int h      = bh & 15;
  const int q_base = blockIdx.x * 16;
  const size_t bh_row0 = (size_t)bh * S_LEN;

  // ---- Phase 1: logits for key columns [wave*256, wave*256+256) ----
  float mqv[8];
  #pragma unroll
  for (int j = 0; j < 8; ++j)
    mqv[j] = mask[b * S_LEN + q_base + half * 8 + j];

  // Hoist the Q fragment (invariant across all 16 key tiles): 16 x v2f.
  const float* qp = q_ws + (bh_row0 + q_base + lr) * D_HEAD + half * 2;
  v2f qfrag[D_HEAD / 4];
  #pragma unroll
  for (int s = 0; s < D_HEAD / 4; ++s)
    qfrag[s] = *(const v2f*)(qp + s * 4);

  for (int t = 0; t < 16; ++t) {
    const int col = wave * 256 + t * 16 + lr;
    const float* kp = k_ws + (bh_row0 + col) * D_HEAD + half * 2;
    v8f acc = {};
    #pragma unroll
    for (int s = 0; s < D_HEAD / 4; ++s) {
      const v2f bfr = *(const v2f*)(kp + s * 4);
      acc = wmma_f32(qfrag[s], bfr, acc);
    }
    const float mk = mask[b * S_LEN + col];
    #pragma unroll
    for (int j = 0; j < 8; ++j) {
      const int rl = half * 8 + j;
      const float logit = acc[j] * SCALE;
      qk_out[(bh_row0 + q_base + rl) * S_LEN + col] = logit;
      const float mval = fminf(1.0f, mqv[j] + mk) * MASK_ALPHA;
      P[rl * LDP + col] = logit + mval;
    }
  }
  __syncthreads();

  // ---- Phase 2: softmax over each of 16 rows (2 rows per wave) ----
  #pragma unroll
  for (int rr = 0; rr < 2; ++rr) {
    const int rl = wave * 2 + rr;
    float* row = P + rl * LDP;
    float mx = -3.0e38f;
    for (int i = lane; i < S_LEN; i += 32) mx = fmaxf(mx, row[i]);
    #pragma unroll
    for (int o = 16; o > 0; o >>= 1) mx = fmaxf(mx, __shfl_xor(mx, o, 32));
    float sum = 0.0f;
    for (int i = lane; i < S_LEN; i += 32) {
      const float e = __expf(row[i] - mx);
      row[i] = e;
      sum += e;
    }
    #pragma unroll
    for (int o = 16; o > 0; o >>= 1) sum += __shfl_xor(sum, o, 32);
    const float inv = 1.0f / sum;
    float* arow = attn_out + (bh_row0 + q_base + rl) * S_LEN;
    for (int i = lane; i < S_LEN; i += 32) {
      const float a = row[i] * inv;
      row[i]  = a;
      arow[i] = a;
    }
  }
  __syncthreads();

  // ---- Phase 3: Z = P(16x2048) @ V(2048x64), split-K across waves ----
  const int ntile = wave & 3;         // which 16 columns of dh
  const int khalf = wave >> 2;        // which half of K
  const int ncol  = ntile * 16 + lr;
  const int kbase = khalf * (S_LEN / 2);

  const float* vp   = v_ws + (bh_row0 + kbase) * D_HEAD + ncol;
  const float* prow = P + lr * LDP + kbase + half * 2;

  v8f acc = {};
  for (int kk = 0; kk < S_LEN / 2; kk += 4) {
    const v2f a = *(const v2f*)(prow + kk);
    v2f bfr;
    bfr.x = vp[(size_t)(kk + half * 2) * D_HEAD];
    bfr.y = vp[(size_t)(kk + half * 2 + 1) * D_HEAD];
    acc = wmma_f32(a, bfr, acc);
  }

  float* rbuf = red + khalf * (16 * RED_LD);
  #pragma unroll
  for (int j = 0; j < 8; ++j)
    rbuf[(half * 8 + j) * RED_LD + ncol] = acc[j];
  __syncthreads();

  if (khalf == 0) {
    #pragma unroll
    for (int j = 0; j < 8; ++j) {
      const int rl = half * 8 + j;
      const float z = red[rl * RED_LD + ncol] +
                      red[16 * RED_LD + rl * RED_LD + ncol];
      z_ws[((size_t)(b * S_LEN + q_base + rl)) * D_MODEL + h * D_HEAD + ncol] = z;
    }
  }
}

// ---------------------------------------------------------------------------
extern "C" void kernel_launch(void* const* d_in, const int* in_sizes, int n_in,
                              void* d_out, int out_size, void* d_ws, size_t ws_size,
                              hipStream_t stream) {
  const float* x    = (const float*)d_in[0];
  const float* mask = (const float*)d_in[1];
  const float* wq   = (const float*)d_in[2];
  const float* bq   = (const float*)d_in[3];
  const float* wk   = (const float*)d_in[4];
  const float* bk   = (const float*)d_in[5];
  const float* wv   = (const float*)d_in[6];
  const float* bv   = (const float*)d_in[7];
  const float* wo   = (const float*)d_in[8];
  const float* bo   = (const float*)d_in[9];

  float* out  = (float*)d_out;                                   // (B,S,D)
  float* attn = out + (size_t)M_ROWS * D_MODEL;                  // (B,H,S,S)
  float* qkv  = attn + (size_t)BATCH * N_HEADS * S_LEN * S_LEN;  // (B,H,S,S)

  float* q_ws = (float*)d_ws;
  float* k_ws = q_ws + (size_t)M_ROWS * D_MODEL;
  float* v_ws = k_ws + (size_t)M_ROWS * D_MODEL;
  float* z_ws = v_ws + (size_t)M_ROWS * D_MODEL;

  const dim3 blk(256, 1, 1);
  const dim3 ggrid(M_ROWS / 128, D_MODEL / 64, 1);

  gemm_proj<<<ggrid, blk, 0, stream>>>(x, wq, bq, q_ws, 1);
  gemm_proj<<<ggrid, blk, 0, stream>>>(x, wk, bk, k_ws, 1);
  gemm_proj<<<ggrid, blk, 0, stream>>>(x, wv, bv, v_ws, 1);

  const size_t smem_bytes =
      (size_t)(16 * LDP + 2 * 16 * RED_LD) * sizeof(float);      // ~135 KB
  const dim3 agrid(S_LEN / 16, BATCH * N_HEADS, 1);
  attn_fused<<<agrid, blk, smem_bytes, stream>>>(q_ws, k_ws, v_ws, mask,
                                                 attn, qkv, z_ws);

  gemm_proj<<<ggrid, blk, 0, stream>>>(z_ws, wo, bo, out, 0);
}